// UnPatcher_5171140624955
// MI455X (gfx1250) — compile-verified
//
#include <hip/hip_runtime.h>

typedef __attribute__((ext_vector_type(2))) float v2f;
typedef __attribute__((ext_vector_type(8))) float v8f;

// Fused 2-level inverse Haar DWT (4x4 patch synthesis), expressed as a
// 16x16x16 f32 GEMM per 16 spatial positions and lowered to
// V_WMMA_F32_16X16X4_F32 chained 4x over K.
//
//   out[b,g,4I+p,4J+q] = sum_{k,m} sign(p,q,k,m) * in[b, g+16k+64m, I, J]
//   sign = (-1)^( (p&1)(k&1) ^ (q&1)(k>>1) ^ (p>>1)(m&1) ^ (q>>1)(m>>1) )
//
// A (16x4 f32, 2 VGPRs): M = lane&15, K = v + 2*(lane>>4)   [ISA 7.12.2]
// B (4x16 f32, 2 VGPRs): N = lane&15, K = v + 2*(lane>>4)   [symmetric]
// C/D (16x16 f32, 8 VGPRs): M = r + 8*(lane>>4), N = lane&15
__global__ __launch_bounds__(256) void idwt2_wmma_kernel(
    const float* __restrict__ x, float* __restrict__ y) {
  const int lane = threadIdx.x & 31;
  const int wid  = (blockIdx.x * 256 + (int)threadIdx.x) >> 5;

  // wid = ((b*16 + g)*96 + I)*6 + jb
  const int jb = wid % 6;
  int tmp = wid / 6;
  const int I = tmp % 96;
  tmp /= 96;
  const int g = tmp & 15;
  const int b = tmp >> 4;

  const int n  = lane & 15;   // A: M index; B/D: N index (spatial j offset)
  const int hi = lane >> 4;   // half-wave: +2 K-offset inside each K=4 chunk
  const int J0 = jb * 16;

  // Sign parameters for A row M = n  (output tile pixel t=n, p=t>>2, q=t&3)
  const int p  = n >> 2, q  = n & 3;
  const int di = p & 1,  dI = (p >> 1) & 1;   // level-2 / level-1 row bits
  const int dj = q & 1,  dJ = (q >> 1) & 1;   // level-2 / level-1 col bits

  // Input base: x[b, 0, I, J0+n]; channel c adds c*96*96 elements.
  const float* xb = x + ((b * 256) * 96 + I) * 96 + (J0 + n);

  v8f acc = {};
#pragma unroll
  for (int ck = 0; ck < 4; ++ck) {
    v2f a, bf;
#pragma unroll
    for (int v = 0; v < 2; ++v) {
      const int s = 4 * ck + v + 2 * hi;          // K index 0..15
      const int k = s & 3, m = s >> 2;            // level-2 / level-1 subband
      const int bit =
          ((di & k) ^ (dj & (k >> 1)) ^ (dI & m) ^ (dJ & (m >> 1))) & 1;
      const float sgn  = bit ? -1.0f : 1.0f;
      const int   c    = g + ((s & 3) << 4) + ((s >> 2) << 6);
      const float bval = xb[c * 9216];            // 9216 = 96*96
      if (v == 0) { a.x = sgn; bf.x = bval; }
      else        { a.y = sgn; bf.y = bval; }
    }
    // D = A*B + C   (f32, full precision; weights are exactly +/-1)
    acc = __builtin_amdgcn_wmma_f32_16x16x4_f32(
        /*neg_a=*/false, a, /*neg_b=*/false, bf,
        /*c_mod=*/(short)0, acc, /*reuse_a=*/false, /*reuse_b=*/false);
  }

  // Lane stores output rows 4I+2*hi and 4I+2*hi+1, cols 4*(J0+n)..+3.
  float* yb = y + ((size_t)((b * 16 + g) * 384 + 4 * I + 2 * hi)) * 384
                + 4 * (J0 + n);
  *(float4*)(yb)       = make_float4(acc[0], acc[1], acc[2], acc[3]);
  *(float4*)(yb + 384) = make_float4(acc[4], acc[5], acc[6], acc[7]);
}

extern "C" void kernel_launch(void* const* d_in, const int* in_sizes, int n_in,
                              void* d_out, int out_size, void* d_ws, size_t ws_size,
                              hipStream_t stream) {
  (void)in_sizes; (void)n_in; (void)out_size; (void)d_ws; (void)ws_size;
  const float* x = (const float*)d_in[0];
  float*       y = (float*)d_out;
  // 8 b * 16 g * 96 I * 6 jblk = 73728 waves = 9216 blocks of 8 waves.
  idwt2_wmma_kernel<<<dim3(9216), dim3(256), 0, stream>>>(x, y);
}